// OrderedAttentionLayer_34359739132
// MI455X (gfx1250) — compile-verified
//
#include <hip/hip_runtime.h>

typedef float v2f __attribute__((ext_vector_type(2)));
typedef float v4f __attribute__((ext_vector_type(4)));
typedef float v8f __attribute__((ext_vector_type(8)));

__device__ __forceinline__ v8f wmma_f32_16x16x4(v2f a, v2f b, v8f c) {
  // D = A(16x4 f32) * B(4x16 f32) + C(16x16 f32)
  return __builtin_amdgcn_wmma_f32_16x16x4_f32(false, a, false, b, (short)0, c, false, false);
}

// ---------------------------------------------------------------------------
// pdense: P = (Wr^T * X * Wc + bias) * (1/16), per 64x64 token matrix.
// transposed==0: store P row-major; transposed==1: store P^T (for v staging).
// ---------------------------------------------------------------------------
struct PDArgs {
  const float* X;   // [192][64][64] token matrices (base; blockIdx.x selects token)
  const float* Wr;  // [64][64]
  const float* Wc;  // [64][64]
  const float* Bb;  // [64][64]
  float* O;         // [192][64][64]
  int transposed;
};
struct PD3 { PDArgs a[3]; };

__global__ __launch_bounds__(128) void pdense_kernel(PD3 P) {
  PDArgs A = P.a[blockIdx.y];
  // stride 68 floats: conflict-free ds reads, 16B-aligned b128 transposed stores
  __shared__ float WcT[64 * 68];
  __shared__ float WrT[64 * 68];
  __shared__ float YT[64 * 68];

  const int tid = threadIdx.x;
  const float* X = A.X + (size_t)blockIdx.x * 4096;

  // cooperative transposed weight staging: WcT[b][j] = Wc[j][b], WrT[a][i] = Wr[i][a]
  for (int idx = tid; idx < 4096; idx += 128) {
    int ri = idx >> 6, ci = idx & 63;
    WcT[ci * 68 + ri] = A.Wc[idx];
    WrT[ci * 68 + ri] = A.Wr[idx];
  }
  __syncthreads();

  const int lane = tid & 31;
  const int mt = tid >> 5;        // wave -> row tile
  const int ln = lane & 15;
  const int h = lane >> 4;

  // Stage B: Y = X @ Wc for rows [mt*16, mt*16+16)
  v8f acc[4] = {};
  for (int ks = 0; ks < 16; ++ks) {
    const int k0 = ks * 4;
    v2f a = *(const v2f*)(X + (mt * 16 + ln) * 64 + k0 + 2 * h);
#pragma unroll
    for (int nt = 0; nt < 4; ++nt) {
      v2f b = *(const v2f*)(&WcT[(nt * 16 + ln) * 68 + k0 + 2 * h]);
      acc[nt] = wmma_f32_16x16x4(a, b, acc[nt]);
    }
  }
  // store Y transposed into LDS: YT[col][row]
#pragma unroll
  for (int nt = 0; nt < 4; ++nt) {
    float* yp = &YT[(nt * 16 + ln) * 68 + mt * 16 + 8 * h];
    *(v4f*)(yp)     = (v4f){acc[nt][0], acc[nt][1], acc[nt][2], acc[nt][3]};
    *(v4f*)(yp + 4) = (v4f){acc[nt][4], acc[nt][5], acc[nt][6], acc[nt][7]};
  }
  __syncthreads();

  // Stage C: P = Wr^T @ Y ; A[M=a][K=i] = WrT[a][i], B^T[b][i] = YT[b][i]
  v8f acc2[4] = {};
  for (int ks = 0; ks < 16; ++ks) {
    const int k0 = ks * 4;
    v2f a = *(const v2f*)(&WrT[(mt * 16 + ln) * 68 + k0 + 2 * h]);
#pragma unroll
    for (int nt = 0; nt < 4; ++nt) {
      v2f b = *(const v2f*)(&YT[(nt * 16 + ln) * 68 + k0 + 2 * h]);
      acc2[nt] = wmma_f32_16x16x4(a, b, acc2[nt]);
    }
  }

  float* O = A.O + (size_t)blockIdx.x * 4096;
#pragma unroll
  for (int nt = 0; nt < 4; ++nt) {
    if (A.transposed) {
      float* op = O + (nt * 16 + ln) * 64 + mt * 16 + 8 * h;
#pragma unroll
      for (int vv = 0; vv < 8; ++vv) {
        float bias = A.Bb[(mt * 16 + vv + 8 * h) * 64 + nt * 16 + ln];
        op[vv] = (acc2[nt][vv] + bias) * 0.0625f;
      }
    } else {
#pragma unroll
      for (int vv = 0; vv < 8; ++vv) {
        int row = mt * 16 + vv + 8 * h, col = nt * 16 + ln;
        float bias = A.Bb[row * 64 + col];
        O[row * 64 + col] = (acc2[nt][vv] + bias) * 0.0625f;
      }
    }
  }
}

// ---------------------------------------------------------------------------
// Attention per (b, l, r) over 16x16 blocks, one wave per query block i.
// qp, kp row-major merged [b*96+t][64][64]; vpT per-token transposed.
// Output ao in merged layout; masked (r>l) blocks written as zeros.
// ---------------------------------------------------------------------------
__global__ __launch_bounds__(64) void attn_kernel(const float* __restrict__ qp,
                                                  const float* __restrict__ kp,
                                                  const float* __restrict__ vpT,
                                                  float* __restrict__ ao) {
  extern __shared__ float smem[];
  const int lane = threadIdx.x & 31;
  const int wv = threadIdx.x >> 5;
  const int i = blockIdx.x * 2 + wv;
  const int lr = blockIdx.y;
  const int l = lr >> 2, r = lr & 3;
  const int b = blockIdx.z;
  const int ln = lane & 15, h = lane >> 4;

  float* outblk = ao + (size_t)(b * 96 + i) * 4096 + (l * 16) * 64 + r * 16;
  if (r > l) {  // post-softmax mask zeroes the whole block (uniform branch)
#pragma unroll
    for (int vv = 0; vv < 8; ++vv) outblk[(vv + 8 * h) * 64 + ln] = 0.0f;
    return;
  }

  // per-wave LDS: 96 exp(S) tiles (stride-20 padded 16x16) + 320-float scratch
  float* Sw = smem + wv * 31040;
  float* Zscr = Sw + 96 * 320;

  // A operand: q block (fold extra 1/16 logits scale into A)
  const float* qb = qp + (size_t)(b * 96 + i) * 4096 + (l * 16 + ln) * 64 + r * 16;
  v2f aq[4];
#pragma unroll
  for (int kk = 0; kk < 4; ++kk) {
    v2f t = *(const v2f*)(qb + 4 * kk + 2 * h);
    aq[kk] = t * 0.0625f;
  }

  // pass 1: S_j = q_i k_j^T, P_j = exp(S_j), Z += P_j, stash P_j in LDS
  v8f zacc = {};
#pragma unroll 2
  for (int j = 0; j < 96; ++j) {
    const float* kb = kp + (size_t)(b * 96 + j) * 4096 + (l * 16 + ln) * 64 + r * 16;
    v8f s = {};
#pragma unroll
    for (int kk = 0; kk < 4; ++kk) {
      v2f bk = *(const v2f*)(kb + 4 * kk + 2 * h);
      s = wmma_f32_16x16x4(aq[kk], bk, s);
    }
    v8f pe;
#pragma unroll
    for (int e = 0; e < 8; ++e) pe[e] = __expf(s[e]);
    zacc += pe;
    float* sp = Sw + j * 320 + ln * 20 + 8 * h;  // column-major [N=m][M=p]
    *(v4f*)(sp)     = (v4f){pe[0], pe[1], pe[2], pe[3]};
    *(v4f*)(sp + 4) = (v4f){pe[4], pe[5], pe[6], pe[7]};
  }

  // Z^-1, transposed D-layout -> A-layout through LDS scratch
  v8f zinv;
#pragma unroll
  for (int e = 0; e < 8; ++e) zinv[e] = 1.0f / zacc[e];
  {
    float* zp = Zscr + ln * 20 + 8 * h;
    *(v4f*)(zp)     = (v4f){zinv[0], zinv[1], zinv[2], zinv[3]};
    *(v4f*)(zp + 4) = (v4f){zinv[4], zinv[5], zinv[6], zinv[7]};
  }
  v2f zA[4];
#pragma unroll
  for (int kk = 0; kk < 4; ++kk) {
    const float* zq = Zscr + (4 * kk + 2 * h) * 20 + ln;
    zA[kk] = (v2f){zq[0], zq[20]};
  }

  // pass 2: O += (P_j / Z) @ v_j
  v8f oacc = {};
#pragma unroll 2
  for (int j = 0; j < 96; ++j) {
    const float* vb = vpT + (size_t)(b * 96 + j) * 4096 + (r * 16 + ln) * 64 + l * 16;
    const float* sj = Sw + j * 320;
#pragma unroll
    for (int kk = 0; kk < 4; ++kk) {
      const float* sp2 = sj + (4 * kk + 2 * h) * 20 + ln;
      v2f w = (v2f){sp2[0], sp2[20]};
      w *= zA[kk];
      v2f bv = *(const v2f*)(vb + 4 * kk + 2 * h);
      oacc = wmma_f32_16x16x4(w, bv, oacc);
    }
  }
#pragma unroll
  for (int vv = 0; vv < 8; ++vv) outblk[(vv + 8 * h) * 64 + ln] = oacc[vv];
}

// ---------------------------------------------------------------------------
extern "C" void kernel_launch(void* const* d_in, const int* in_sizes, int n_in,
                              void* d_out, int out_size, void* d_ws, size_t ws_size,
                              hipStream_t stream) {
  (void)in_sizes; (void)n_in; (void)out_size; (void)ws_size;
  const float* Q      = (const float*)d_in[0];
  const float* Kin    = (const float*)d_in[1];
  const float* Vin    = (const float*)d_in[2];
  const float* Wq_row = (const float*)d_in[3];
  const float* Wq_col = (const float*)d_in[4];
  const float* bq     = (const float*)d_in[5];
  const float* Wk_row = (const float*)d_in[6];
  const float* Wk_col = (const float*)d_in[7];
  const float* bk     = (const float*)d_in[8];
  const float* Wv_row = (const float*)d_in[9];
  const float* Wv_col = (const float*)d_in[10];
  const float* bv     = (const float*)d_in[11];
  const float* Wo_row = (const float*)d_in[12];
  const float* Wo_col = (const float*)d_in[13];
  const float* bo     = (const float*)d_in[14];

  float* ws = (float*)d_ws;
  const size_t TOKF = (size_t)192 * 4096;  // one [192][64][64] tensor
  float* qpw = ws;
  float* kpw = ws + TOKF;
  float* vpt = ws + 2 * TOKF;
  float* aow = ws + 3 * TOKF;

  PD3 p3;
  p3.a[0] = PDArgs{Q,   Wq_row, Wq_col, bq, qpw, 0};
  p3.a[1] = PDArgs{Kin, Wk_row, Wk_col, bk, kpw, 0};
  p3.a[2] = PDArgs{Vin, Wv_row, Wv_col, bv, vpt, 1};
  hipLaunchKernelGGL(pdense_kernel, dim3(192, 3), dim3(128), 0, stream, p3);

  const size_t shmem = (size_t)2 * 31040 * sizeof(float);  // 2 waves * (96*320+320)
  hipLaunchKernelGGL(attn_kernel, dim3(48, 16, 2), dim3(64), shmem, stream,
                     qpw, kpw, vpt, aow);

  PD3 p1;
  p1.a[0] = PDArgs{aow, Wo_row, Wo_col, bo, (float*)d_out, 0};
  p1.a[1] = p1.a[0];
  p1.a[2] = p1.a[0];
  hipLaunchKernelGGL(pdense_kernel, dim3(192, 1), dim3(128), 0, stream, p1);
}